// SpatialMLP_15479062135087
// MI455X (gfx1250) — compile-verified
//
#include <hip/hip_runtime.h>

// Problem constants (from reference)
#define B_      16
#define N_IN    196608
#define C_INP   16
#define N_OUTT  49152
#define KNB     4
#define H1_     64
#define H2_     64
#define H3_     32

// LDS strides (floats)
#define XS    68    // x/h tile row stride: A-fragment b64 reads conflict-free
#define WSP   160   // paired-weight K-pair stride (W1/W2): 160 mod 64 = 32 -> halves disjoint
#define WSP3  96    // paired-weight K-pair stride (W3):     96 mod 64 = 32

typedef __attribute__((ext_vector_type(2))) float v2f;
typedef __attribute__((ext_vector_type(8))) float v8f;

__device__ __forceinline__ float gelu_f(float x) {
    float x3 = x * x * x;
    float y  = 0.7978845608028654f * (x + 0.044715f * x3);
#if __has_builtin(__builtin_amdgcn_tanhf)
    float t = __builtin_amdgcn_tanhf(y);     // v_tanh_f32 (gfx1250 trans op)
#else
    float t = tanhf(y);
#endif
    return 0.5f * x * (1.0f + t);
}

__global__ __launch_bounds__(256)
void spatial_mlp_wmma_f32(const float* __restrict__ x,
                          const int*   __restrict__ conn,
                          const float* __restrict__ W1, const float* __restrict__ b1,
                          const float* __restrict__ W2, const float* __restrict__ b2,
                          const float* __restrict__ W3, const float* __restrict__ b3,
                          float* __restrict__ out)
{
    // 86 KB total LDS
    __shared__ __align__(16) float xt [128 * XS];       // 34816 B gathered rows / h scratch
    __shared__ __align__(16) float w1p[32 * WSP];       // 20480 B  pairs: {W[2p][n],W[2p+1][n]}
    __shared__ __align__(16) float w2p[32 * WSP];       // 20480 B
    __shared__ __align__(16) float w3p[32 * WSP3];      // 12288 B

    const int tid   = threadIdx.x;
    const int blk   = blockIdx.x;                 // 6144 blocks
    const int bIdx  = blk / (N_OUTT / 128);       // batch index
    const int nBase = (blk % (N_OUTT / 128)) * 128;

    // ---- stage weights into LDS, K-pair interleaved ----
    for (int i = tid; i < H1_ * H1_; i += 256) {
        int k = i >> 6, n = i & 63;
        w1p[(k >> 1) * WSP + 2 * n + (k & 1)] = W1[i];
    }
    for (int i = tid; i < H1_ * H2_; i += 256) {
        int k = i >> 6, n = i & 63;
        w2p[(k >> 1) * WSP + 2 * n + (k & 1)] = W2[i];
    }
    for (int i = tid; i < H2_ * H3_; i += 256) {
        int k = i >> 5, n = i & 31;
        w3p[(k >> 1) * WSP3 + 2 * n + (k & 1)] = W3[i];
    }

    // ---- gather: 128 rows x 4 neighbors x 16 floats (64B chunks) ----
    for (int c = tid; c < 128 * KNB; c += 256) {
        int row = c >> 2, j = c & 3;
        int idx = conn[(nBase + row) * KNB + j];
        const float4* src = (const float4*)(x + ((long long)bIdx * N_IN + idx) * C_INP);
        float4 v0 = src[0], v1 = src[1], v2 = src[2], v3 = src[3];
        float* dst = &xt[row * XS + j * C_INP];
        ((float4*)dst)[0] = v0; ((float4*)dst)[1] = v1;
        ((float4*)dst)[2] = v2; ((float4*)dst)[3] = v3;
    }
    __syncthreads();

    // ---- per-wave 16-row tile through the MLP ----
    const int w    = tid >> 5;       // wave id 0..7
    const int lane = tid & 31;
    const int l15  = lane & 15;
    const int hi   = lane >> 4;      // half-wave: K-local {0,1} vs {2,3}
    float* tile = &xt[w * 16 * XS];

    // Per-lane base pointers: all per-step offsets become static immediates
    const float* aB  = &tile[l15 * XS + 2 * hi];          // A frag: + 4*kk
    const float* w1B = &w1p[hi * WSP  + 2 * l15];         // B frag: + 2*kk*WSP  + 2*n0
    const float* w2B = &w2p[hi * WSP  + 2 * l15];
    const float* w3B = &w3p[hi * WSP3 + 2 * l15];
    float*       sB  = &tile[8 * hi * XS + l15];          // D store: + v*XS + n0

    // ======== Layer 1: 16x64 @ 64x64, GELU ========
    {
        v8f a0 = {}, a1 = {}, a2 = {}, a3 = {};
#pragma unroll
        for (int kk = 0; kk < 16; ++kk) {
            v2f av  = *(const v2f*)(aB  + 4 * kk);
            const float* bp = w1B + 2 * kk * WSP;
            v2f bw0 = *(const v2f*)(bp +  0);
            v2f bw1 = *(const v2f*)(bp + 32);
            v2f bw2 = *(const v2f*)(bp + 64);
            v2f bw3 = *(const v2f*)(bp + 96);
            a0 = __builtin_amdgcn_wmma_f32_16x16x4_f32(false, av, false, bw0, (short)0, a0, false, false);
            a1 = __builtin_amdgcn_wmma_f32_16x16x4_f32(false, av, false, bw1, (short)0, a1, false, false);
            a2 = __builtin_amdgcn_wmma_f32_16x16x4_f32(false, av, false, bw2, (short)0, a2, false, false);
            a3 = __builtin_amdgcn_wmma_f32_16x16x4_f32(false, av, false, bw3, (short)0, a3, false, false);
        }
        float bb0 = b1[ 0 + l15], bb1 = b1[16 + l15], bb2 = b1[32 + l15], bb3 = b1[48 + l15];
#pragma unroll
        for (int v = 0; v < 8; ++v) {
            sB[v * XS +  0] = gelu_f(a0[v] + bb0);
            sB[v * XS + 16] = gelu_f(a1[v] + bb1);
            sB[v * XS + 32] = gelu_f(a2[v] + bb2);
            sB[v * XS + 48] = gelu_f(a3[v] + bb3);
        }
        asm volatile("s_wait_dscnt 0" ::: "memory");
    }

    // ======== Layer 2: 16x64 @ 64x64, GELU ========
    {
        v8f a0 = {}, a1 = {}, a2 = {}, a3 = {};
#pragma unroll
        for (int kk = 0; kk < 16; ++kk) {
            v2f av  = *(const v2f*)(aB  + 4 * kk);
            const float* bp = w2B + 2 * kk * WSP;
            v2f bw0 = *(const v2f*)(bp +  0);
            v2f bw1 = *(const v2f*)(bp + 32);
            v2f bw2 = *(const v2f*)(bp + 64);
            v2f bw3 = *(const v2f*)(bp + 96);
            a0 = __builtin_amdgcn_wmma_f32_16x16x4_f32(false, av, false, bw0, (short)0, a0, false, false);
            a1 = __builtin_amdgcn_wmma_f32_16x16x4_f32(false, av, false, bw1, (short)0, a1, false, false);
            a2 = __builtin_amdgcn_wmma_f32_16x16x4_f32(false, av, false, bw2, (short)0, a2, false, false);
            a3 = __builtin_amdgcn_wmma_f32_16x16x4_f32(false, av, false, bw3, (short)0, a3, false, false);
        }
        float bb0 = b2[ 0 + l15], bb1 = b2[16 + l15], bb2 = b2[32 + l15], bb3 = b2[48 + l15];
#pragma unroll
        for (int v = 0; v < 8; ++v) {
            sB[v * XS +  0] = gelu_f(a0[v] + bb0);
            sB[v * XS + 16] = gelu_f(a1[v] + bb1);
            sB[v * XS + 32] = gelu_f(a2[v] + bb2);
            sB[v * XS + 48] = gelu_f(a3[v] + bb3);
        }
        asm volatile("s_wait_dscnt 0" ::: "memory");
    }

    // ======== Layer 3: 16x64 @ 64x32, bias only, store to global ========
    {
        v8f c0 = {}, c1 = {};
#pragma unroll
        for (int kk = 0; kk < 16; ++kk) {
            v2f av  = *(const v2f*)(aB  + 4 * kk);
            const float* bp = w3B + 2 * kk * WSP3;
            v2f bw0 = *(const v2f*)(bp +  0);
            v2f bw1 = *(const v2f*)(bp + 32);
            c0 = __builtin_amdgcn_wmma_f32_16x16x4_f32(false, av, false, bw0, (short)0, c0, false, false);
            c1 = __builtin_amdgcn_wmma_f32_16x16x4_f32(false, av, false, bw1, (short)0, c1, false, false);
        }
        float bo0 = b3[l15], bo1 = b3[16 + l15];
        long long rowBase = (long long)bIdx * N_OUTT + nBase + w * 16 + 8 * hi;
        float* outB = out + rowBase * H3_ + l15;
#pragma unroll
        for (int v = 0; v < 8; ++v) {
            outB[v * H3_ +  0] = c0[v] + bo0;
            outB[v * H3_ + 16] = c1[v] + bo1;
        }
    }
}

extern "C" void kernel_launch(void* const* d_in, const int* in_sizes, int n_in,
                              void* d_out, int out_size, void* d_ws, size_t ws_size,
                              hipStream_t stream) {
    (void)in_sizes; (void)n_in; (void)out_size; (void)d_ws; (void)ws_size;
    const float* x    = (const float*)d_in[0];
    const int*   conn = (const int*)  d_in[1];
    const float* W1   = (const float*)d_in[2];
    const float* b1   = (const float*)d_in[3];
    const float* W2   = (const float*)d_in[4];
    const float* b2   = (const float*)d_in[5];
    const float* W3   = (const float*)d_in[6];
    const float* b3   = (const float*)d_in[7];
    float* out = (float*)d_out;

    const int blocks = (B_ * N_OUTT) / 128;   // 6144
    spatial_mlp_wmma_f32<<<blocks, 256, 0, stream>>>(x, conn, W1, b1, W2, b2, W3, b3, out);
}